// SpeculativeDecoder_45157286150326
// MI455X (gfx1250) — compile-verified
//
#include <hip/hip_runtime.h>
#include <hip/hip_bf16.h>
#include <math.h>

// Speculative decoding verification for MI455X (gfx1250, wave32).
// Bandwidth-bound: 257 MB logits streamed once (kernel 1) + ~51 MB fallback
// pass (kernel 3).  Block reductions use V_WMMA_F32_16X16X4_F32 with a ones
// B-matrix (exact f32 row-sum accumulation), lane-permutation-invariant since
// we sum every element of every A tile.

#define PAD_IDX 0
#define KSPEC 5
#define BATCH 256
#define VOCAB 50257
#define V4 (VOCAB / 4)          // 12564 float4s, 1 tail element
#define NTHREADS 256
#define SEG 197                 // ceil(VOCAB / 256); 255*197 = 50235 < VOCAB

typedef float v2f __attribute__((ext_vector_type(2)));
typedef float v8f __attribute__((ext_vector_type(8)));

// Sum 256 f32 values sitting in LDS. Must be executed by all 32 lanes of one
// wave (EXEC all ones -- WMMA requirement). Four accumulating
// v_wmma_f32_16x16x4_f32 with B == ones(4x16) produce C[m][n] = sum of A-row m
// over all tiles; summing the 8 C VGPRs gives the half-matrix sum per lane,
// and one shfl_xor(16) combines the two halves. Total lands in every lane.
__device__ __forceinline__ float wmma_reduce256(const float* vals) {
  int lane = threadIdx.x & 31;
  v8f c = {0.f, 0.f, 0.f, 0.f, 0.f, 0.f, 0.f, 0.f};
  v2f ones;
  ones.x = 1.f;
  ones.y = 1.f;
#pragma unroll
  for (int i = 0; i < 4; ++i) {
    v2f a;
    a.x = vals[i * 64 + lane * 2 + 0];
    a.y = vals[i * 64 + lane * 2 + 1];
    // (neg_a, A, neg_b, B, c_mod, C, reuse_a, reuse_b)
    c = __builtin_amdgcn_wmma_f32_16x16x4_f32(false, a, false, ones,
                                              (short)0, c, false, false);
  }
  float r = c[0] + c[1] + c[2] + c[3] + c[4] + c[5] + c[6] + c[7];
  r += __shfl_xor(r, 16, 32);
  return r;
}

// ---------------- Kernel 1: per-(b,k) row logsumexp + draft-token gather ----
// One block per row; online softmax so the 257 MB tensor is streamed exactly
// once (roofline: ~11 us at 23.3 TB/s).
__global__ void k_logsumexp(const float* __restrict__ logits,
                            const int* __restrict__ draft_tokens,
                            float* __restrict__ ws_tlp,
                            float* __restrict__ ws_logZ) {
  __shared__ float sm[NTHREADS];
  __shared__ float ss[NTHREADS];
  int row = blockIdx.x;   // 0 .. BATCH*KSPEC-1
  int t = threadIdx.x;
  const float* x = logits + (size_t)row * VOCAB;
  const float4* x4 = (const float4*)x;

  float m = -INFINITY, s = 0.f;
  for (int i = t; i < V4; i += NTHREADS) {
    float4 v = x4[i];
    __builtin_prefetch(&x4[i + NTHREADS], 0, 0);   // global_prefetch_b8
    float cmax = fmaxf(fmaxf(v.x, v.y), fmaxf(v.z, v.w));
    if (cmax > m) { s *= __expf(m - cmax); m = cmax; }
    s += __expf(v.x - m) + __expf(v.y - m) + __expf(v.z - m) + __expf(v.w - m);
  }
  if (t == 0) {  // tail element (VOCAB = 4*V4 + 1)
    float v = x[VOCAB - 1];
    if (v > m) { s *= __expf(m - v); m = v; }
    s += __expf(v - m);
  }

  // block max
  sm[t] = m;
  __syncthreads();
  for (int off = NTHREADS / 2; off > 0; off >>= 1) {
    if (t < off) sm[t] = fmaxf(sm[t], sm[t + off]);
    __syncthreads();
  }
  float M = sm[0];
  __syncthreads();

  // rescale partial sums to the common max, then WMMA block-sum
  ss[t] = s * __expf(m - M);
  __syncthreads();
  if (t < 32) {
    float S = wmma_reduce256(ss);
    if (t == 0) {
      float logZ = M + __logf(S);
      ws_logZ[row] = logZ;
      int tok = draft_tokens[row];
      ws_tlp[row] = x[tok] - logZ;   // log_softmax at the draft token
    }
  }
}

// ---------------- Kernel 2: accept/reject scan (tiny) -----------------------
__global__ void k_accept(const int* __restrict__ draft_tokens,
                         const float* __restrict__ draft_log_probs,
                         const float* __restrict__ random_vals,
                         const float* __restrict__ ws_tlp,
                         float* __restrict__ out_accepted,  // [B*K] as float
                         float* __restrict__ out_n,         // [B] as float
                         int* __restrict__ ws_fr) {
  int b = blockIdx.x * blockDim.x + threadIdx.x;
  if (b >= BATCH) return;
  int fr = KSPEC;
  for (int k = 0; k < KSPEC; ++k) {
    float tlp = ws_tlp[b * KSPEC + k];
    float dlp = draft_log_probs[b * KSPEC + k];
    float ap = fminf(__expf(tlp - dlp), 1.f);
    bool acc = random_vals[b * KSPEC + k] < ap;
    if (!acc && fr == KSPEC) fr = k;
  }
  for (int k = 0; k < KSPEC; ++k)
    out_accepted[b * KSPEC + k] =
        (k < fr) ? (float)draft_tokens[b * KSPEC + k] : (float)PAD_IDX;
  out_n[b] = (float)fr;
  ws_fr[b] = fr;
}

// ---------------- Kernel 3: fallback sampling via segmented inverse-CDF -----
// One block per batch row. Per-thread contiguous segment keeps vocab order so
// the inclusive scan over 256 segment masses locates the CDF crossing; only
// the owning thread re-walks its 197 elements.
__global__ void k_fallback(const float* __restrict__ logits,
                           const float* __restrict__ draft_log_probs,
                           const float* __restrict__ uniform_sample,
                           const float* __restrict__ ws_logZ,
                           const int* __restrict__ ws_fr,
                           float* __restrict__ out_final) {
  __shared__ float la[NTHREADS];
  __shared__ float lp[NTHREADS];
  __shared__ float scan[NTHREADS];
  __shared__ float bc[2];
  __shared__ int result;

  int b = blockIdx.x;
  int t = threadIdx.x;
  int fr = ws_fr[b];
  int rp = (fr < KSPEC) ? fr : (KSPEC - 1);
  int row = b * KSPEC + rp;
  const float* x = logits + (size_t)row * VOCAB;
  float logZ = ws_logZ[row];
  float d = __expf(draft_log_probs[row]);

  int start = t * SEG;
  int end = start + SEG;
  if (end > VOCAB) end = VOCAB;

  float accA = 0.f, accP = 0.f;
  for (int i = start; i < end; ++i) {
    float p = __expf(x[i] - logZ);
    accP += p;
    accA += fmaxf(p - d, 0.f);
  }
  la[t] = accA;
  lp[t] = accP;
  if (t == 0) result = VOCAB - 1;
  __syncthreads();

  if (t < 32) {
    float Sa = wmma_reduce256(la);
    float Sp = wmma_reduce256(lp);
    if (t == 0) { bc[0] = Sa; bc[1] = Sp; }
  }
  __syncthreads();

  float Sadj = bc[0];
  bool use_adj = Sadj > 0.f;
  float invS = use_adj ? 1.f / Sadj : 1.f;
  float c = use_adj ? accA * invS : accP;   // this thread's probability mass

  // Hillis-Steele inclusive scan over the 256 segment masses
  scan[t] = c;
  __syncthreads();
  float incl = c;
  for (int off = 1; off < NTHREADS; off <<= 1) {
    float add = (t >= off) ? scan[t - off] : 0.f;
    __syncthreads();
    incl += add;
    scan[t] = incl;
    __syncthreads();
  }
  float total = scan[NTHREADS - 1];
  float thresh = uniform_sample[b] * total;

  // unique owner: smallest t whose inclusive prefix crosses thresh
  bool owner = (incl >= thresh) && (t == 0 || scan[t - 1] < thresh);
  if (owner) {
    float running = incl - c;   // exclusive prefix
    int tok = end - 1;          // in-segment fallback against rounding
    for (int i = start; i < end; ++i) {
      float p = __expf(x[i] - logZ);
      float pr = use_adj ? fmaxf(p - d, 0.f) * invS : p;
      running += pr;
      if (running >= thresh) { tok = i; break; }
    }
    result = tok;
  }
  __syncthreads();
  if (t == 0) out_final[b] = (fr < KSPEC) ? (float)result : (float)PAD_IDX;
}

extern "C" void kernel_launch(void* const* d_in, const int* in_sizes, int n_in,
                              void* d_out, int out_size, void* d_ws,
                              size_t ws_size, hipStream_t stream) {
  const int* draft_tokens = (const int*)d_in[0];        // [B,K] int32
  const float* draft_log_probs = (const float*)d_in[1]; // [B,K]
  const float* target_logits = (const float*)d_in[2];   // [B,K,V]
  const float* random_vals = (const float*)d_in[3];     // [B,K]
  const float* uniform_sample = (const float*)d_in[4];  // [B]

  float* out = (float*)d_out;
  float* out_accepted = out;                  // B*K = 1280
  float* out_n = out + BATCH * KSPEC;         // 256
  float* out_final = out_n + BATCH;           // 256

  float* ws_tlp = (float*)d_ws;               // B*K
  float* ws_logZ = ws_tlp + BATCH * KSPEC;    // B*K
  int* ws_fr = (int*)(ws_logZ + BATCH * KSPEC); // B
  (void)in_sizes; (void)n_in; (void)out_size; (void)ws_size;

  k_logsumexp<<<BATCH * KSPEC, NTHREADS, 0, stream>>>(
      target_logits, draft_tokens, ws_tlp, ws_logZ);
  k_accept<<<1, NTHREADS, 0, stream>>>(
      draft_tokens, draft_log_probs, random_vals, ws_tlp,
      out_accepted, out_n, ws_fr);
  k_fallback<<<BATCH, NTHREADS, 0, stream>>>(
      target_logits, draft_log_probs, uniform_sample, ws_logZ, ws_fr,
      out_final);
}